// GCN_63015760167230
// MI455X (gfx1250) — compile-verified
//
#include <hip/hip_runtime.h>
#include <hip/hip_bf16.h>
#include <stdint.h>

#define F_IN  128
#define F_HID 128
#define F_OUT 64
#define EPS_BN 1e-5f

typedef __attribute__((ext_vector_type(2))) float v2f;
typedef __attribute__((ext_vector_type(8))) float v8f;

// ---------------- degree / normalization ----------------

__global__ void fill_ones_kernel(float* __restrict__ deg, int n) {
  int i = blockIdx.x * blockDim.x + threadIdx.x;
  if (i < n) deg[i] = 1.0f;  // self-loop contributes 1 to every node's degree
}

__global__ void deg_accum_kernel(const long long* __restrict__ col,
                                 float* __restrict__ deg, int e) {
  int i = blockIdx.x * blockDim.x + threadIdx.x;
  if (i < e) atomicAdd(&deg[(int)col[i]], 1.0f);
}

__global__ void rsqrt_inplace_kernel(float* __restrict__ deg, int n) {
  int i = blockIdx.x * blockDim.x + threadIdx.x;
  if (i < n) {
    float d = deg[i];
    deg[i] = (d > 0.0f) ? rsqrtf(d) : 0.0f;
  }
}

// ---------------- fp32 WMMA GEMM:  C[N x FOUT] = H[N x 128] @ W[128 x FOUT] ----------------
// Block: FOUT/16 waves; wave w computes the 16x16 tile (rows m0..m0+15, cols 16w..16w+15).
// A-tile (16 rows x 128) staged in LDS with +4 float pad -> conflict-free fragment reads.
// Fragment layouts per CDNA5 ISA 7.12.2 (32-bit A 16x4, B 4x16, C/D 16x16).

template <int FOUT>
__global__ void __launch_bounds__(FOUT * 2)
gemm_wmma_f32_kernel(const float* __restrict__ H, const float* __restrict__ W,
                     float* __restrict__ C, int nRows) {
  constexpr int K  = 128;
  constexpr int NW = FOUT / 16;           // waves per block
  __shared__ float As[16][K + 4];         // row stride 132 floats (== 4 mod 64 banks)

  const int m0 = blockIdx.x * 16;
  if (m0 >= nRows) return;                // uniform per block: EXEC stays all-ones

  const int t = threadIdx.x;
  // cooperative, coalesced load of the A tile (16*128 floats)
  for (int idx = t; idx < 16 * K; idx += NW * 32) {
    const int r = idx >> 7;
    const int k = idx & (K - 1);
    As[r][k] = H[(size_t)(m0 + r) * K + k];
  }
  __syncthreads();

  const int wave  = t >> 5;
  const int lane  = t & 31;
  const int mrow  = lane & 15;            // A: M = lane%16
  const int khalf = (lane >> 4) << 1;     // lanes 16..31 hold K+2, K+3
  const int n     = (wave << 4) + mrow;   // B/C/D: N = lane%16 within this wave's tile

  v8f acc = {};
#pragma unroll
  for (int k0 = 0; k0 < K; k0 += 4) {
    const int ka = k0 + khalf;
    v2f a;
    a.x = As[mrow][ka];
    a.y = As[mrow][ka + 1];
    v2f b;
    b.x = W[(size_t)ka * FOUT + n];
    b.y = W[(size_t)(ka + 1) * FOUT + n];
    // 8 args: (neg_a, A, neg_b, B, c_mod, C, reuse_a, reuse_b)
    acc = __builtin_amdgcn_wmma_f32_16x16x4_f32(false, a, false, b,
                                                (short)0, acc, false, false);
  }

  // D layout: lanes 0-15 -> rows v (v=0..7), lanes 16-31 -> rows v+8
  const int mBase = m0 + ((lane >> 4) << 3);
#pragma unroll
  for (int v = 0; v < 8; ++v)
    C[(size_t)(mBase + v) * FOUT + n] = acc[v];
}

// ---------------- self-loop init:  agg = dinv[i]^2 * hw + bias ----------------

template <int F>
__global__ void agg_init_kernel(const float* __restrict__ hw,
                                const float* __restrict__ dinv,
                                const float* __restrict__ bias,
                                float* __restrict__ agg, int n) {
  const long long idx = (long long)blockIdx.x * blockDim.x + threadIdx.x;
  if (idx >= (long long)n * F) return;
  const int i = (int)(idx / F);
  const int f = (int)(idx - (long long)i * F);
  const float d = dinv[i];
  agg[idx] = d * d * hw[idx] + bias[f];
}

// ---------------- edge scatter:  agg[col] += dinv[row]*dinv[col] * hw[row] ----------------
// One wave per edge; lane covers F/32 consecutive floats -> coalesced gather,
// global_atomic_add_f32 scatter (mostly L2-resident: 51 MB feature matrix << 192 MB L2).

template <int F>
__global__ void scatter_kernel(const float* __restrict__ hw,
                               const long long* __restrict__ row,
                               const long long* __restrict__ col,
                               const float* __restrict__ dinv,
                               float* agg, int e) {
  const int gw = (int)(((long long)blockIdx.x * blockDim.x + threadIdx.x) >> 5);
  if (gw >= e) return;
  const int lane = threadIdx.x & 31;
  const int r = (int)row[gw];
  const int c = (int)col[gw];
  const float nrm = dinv[r] * dinv[c];

  constexpr int V = F / 32;
  const float* src = hw + (size_t)r * F + lane * V;
  float* dst = agg + (size_t)c * F + lane * V;
#pragma unroll
  for (int i = 0; i < V; ++i) atomicAdd(dst + i, nrm * src[i]);
}

// ---------------- fused batchnorm (inference) + relu, in place ----------------

__global__ void bn_relu_kernel(float* __restrict__ h,
                               const float* __restrict__ g, const float* __restrict__ be,
                               const float* __restrict__ m, const float* __restrict__ v,
                               long long total) {
  const long long idx = (long long)blockIdx.x * blockDim.x + threadIdx.x;
  if (idx >= total) return;
  const int f = (int)(idx & (F_HID - 1));
  float x = h[idx];
  x = (x - m[f]) * rsqrtf(v[f] + EPS_BN) * g[f] + be[f];
  h[idx] = x > 0.0f ? x : 0.0f;
}

// ---------------- launch ----------------

extern "C" void kernel_launch(void* const* d_in, const int* in_sizes, int n_in,
                              void* d_out, int out_size, void* d_ws, size_t ws_size,
                              hipStream_t stream) {
  (void)n_in; (void)out_size; (void)ws_size;

  const float*     x   = (const float*)d_in[0];
  const long long* ei  = (const long long*)d_in[1];   // int64 per reference, shape (2, E)
  const float*     W1  = (const float*)d_in[2];
  const float*     b1  = (const float*)d_in[3];
  const float*     W2  = (const float*)d_in[4];
  const float*     b2  = (const float*)d_in[5];
  const float*     W3  = (const float*)d_in[6];
  const float*     b3  = (const float*)d_in[7];
  const float*     g1  = (const float*)d_in[8];
  const float*     be1 = (const float*)d_in[9];
  const float*     m1  = (const float*)d_in[10];
  const float*     v1  = (const float*)d_in[11];
  const float*     g2  = (const float*)d_in[12];
  const float*     be2 = (const float*)d_in[13];
  const float*     m2  = (const float*)d_in[14];
  const float*     v2  = (const float*)d_in[15];

  const int n = in_sizes[0] / F_IN;   // 100000
  const int e = in_sizes[1] / 2;      // 1600000
  const long long* row = ei;          // source nodes
  const long long* col = ei + e;      // target nodes

  // workspace layout
  char*  ws  = (char*)d_ws;
  size_t off = 0;
  auto carve = [&](size_t bytes) -> void* {
    void* p = ws + off;
    off += (bytes + 255) & ~(size_t)255;
    return p;
  };
  float* dinv = (float*)carve((size_t)n * sizeof(float));          // deg -> dinv in place
  float* bufA = (float*)carve((size_t)n * F_HID * sizeof(float));  // h @ W
  float* bufB = (float*)carve((size_t)n * F_HID * sizeof(float));  // agg / activations

  const int T = 256;
  const int nBlk  = (n + T - 1) / T;
  const int eBlk  = (e + T - 1) / T;
  const int mTiles = (n + 15) / 16;                       // 6250 exact
  const long long elemsH = (long long)n * F_HID;
  const long long elemsO = (long long)n * F_OUT;
  const int hBlk = (int)((elemsH + T - 1) / T);
  const int oBlk = (int)((elemsO + T - 1) / T);
  const int ewBlk = (int)(((long long)e * 32 + T - 1) / T);  // 1 wave per edge

  // --- GCN normalization: deg = 1 + sum_in, dinv = deg^-1/2 ---
  fill_ones_kernel<<<nBlk, T, 0, stream>>>(dinv, n);
  deg_accum_kernel<<<eBlk, T, 0, stream>>>(col, dinv, e);
  rsqrt_inplace_kernel<<<nBlk, T, 0, stream>>>(dinv, n);

  // --- layer 1: relu(bn(conv(x, W1, b1))) ---
  gemm_wmma_f32_kernel<F_HID><<<mTiles, F_HID * 2, 0, stream>>>(x, W1, bufA, n);
  agg_init_kernel<F_HID><<<hBlk, T, 0, stream>>>(bufA, dinv, b1, bufB, n);
  scatter_kernel<F_HID><<<ewBlk, T, 0, stream>>>(bufA, row, col, dinv, bufB, e);
  bn_relu_kernel<<<hBlk, T, 0, stream>>>(bufB, g1, be1, m1, v1, elemsH);

  // --- layer 2: relu(bn(conv(h1, W2, b2))) ---
  gemm_wmma_f32_kernel<F_HID><<<mTiles, F_HID * 2, 0, stream>>>(bufB, W2, bufA, n);
  agg_init_kernel<F_HID><<<hBlk, T, 0, stream>>>(bufA, dinv, b2, bufB, n);
  scatter_kernel<F_HID><<<ewBlk, T, 0, stream>>>(bufA, row, col, dinv, bufB, e);
  bn_relu_kernel<<<hBlk, T, 0, stream>>>(bufB, g2, be2, m2, v2, elemsH);

  // --- layer 3: conv(h2, W3, b3) -> d_out ---
  gemm_wmma_f32_kernel<F_OUT><<<mTiles, F_OUT * 2, 0, stream>>>(bufB, W3, bufA, n);
  agg_init_kernel<F_OUT><<<oBlk, T, 0, stream>>>(bufA, dinv, b3, (float*)d_out, n);
  scatter_kernel<F_OUT><<<ewBlk, T, 0, stream>>>(bufA, row, col, dinv, (float*)d_out, e);
}